// GraphConvolution_28157805592769
// MI455X (gfx1250) — compile-verified
//
#include <hip/hip_runtime.h>
#include <hip/hip_bf16.h>

typedef __attribute__((ext_vector_type(2))) float v2f;
typedef __attribute__((ext_vector_type(8))) float v8f;

#define IN_F 256
#define OUT_F 128

// -------------------------------------------------------------------------
// Kernel: scatter-add (dominant phase). One wave per edge:
//   - edge index/value forced into SGPRs via readfirstlane -> s_load_b32
//   - each lane gathers float4 of the source row (global_load_b128,
//     512B/wave coalesced; support lives in 192MB L2)
//   - 4x global_atomic_add_f32 (non-returning -> STOREcnt, resolves in L2)
// Placed first in the file so the disasm snippet shows its lowering.
// -------------------------------------------------------------------------
__global__ void __launch_bounds__(256)
gcn_scatter(const float* __restrict__ support,
            const int* __restrict__ erow, const int* __restrict__ ecol,
            const float* __restrict__ eval, float* __restrict__ out, int E) {
  const int wave = (int)((blockIdx.x * 256u + threadIdx.x) >> 5);
  const int lane = (int)(threadIdx.x & 31u);
  const int e = __builtin_amdgcn_readfirstlane(wave);  // wave-uniform -> SGPR
  if (e >= E) return;                                   // uniform branch
  const int   src = ecol[e];                            // s_load_b32
  const int   dst = erow[e];                            // s_load_b32
  const float val = eval[e];                            // s_load_b32

  const float4 s = *(const float4*)(support + (size_t)src * OUT_F + lane * 4);
  float* __restrict__ o = out + (size_t)dst * OUT_F + lane * 4;
  unsafeAtomicAdd(o + 0, s.x * val);
  unsafeAtomicAdd(o + 1, s.y * val);
  unsafeAtomicAdd(o + 2, s.z * val);
  unsafeAtomicAdd(o + 3, s.w * val);
}

// -------------------------------------------------------------------------
// Kernel: support = x @ W using V_WMMA_F32_16X16X4_F32 (exact fp32 path).
// One wave computes a 16x128 slab: 8 N-tiles of 16x16, K-loop 256 in steps
// of 4. A-fragment layout (ISA 7.12.2, 32-bit A 16x4): lanes 0-15 hold
// K=0,1 in v0,v1; lanes 16-31 hold K=2,3. B mirrored (4x16, row striped
// across lanes). C/D: lane L -> col = L&15, rows m = (L>>4)*8 + j.
// -------------------------------------------------------------------------
__global__ void __launch_bounds__(256)
gcn_gemm_wmma(const float* __restrict__ x, const float* __restrict__ w,
              float* __restrict__ support, int M, int mtiles) {
  const int wave  = (int)((blockIdx.x * (unsigned)blockDim.x + threadIdx.x) >> 5);
  const int lane  = (int)(threadIdx.x & 31u);
  const int mtile = wave;
  if (mtile >= mtiles) return;                  // uniform per wave: EXEC stays all-1s

  const int halfsel = lane >> 4;                // 0 or 1
  const int sub     = lane & 15;                // row (A) / col (B,C)
  const int khalf   = halfsel << 1;             // 0 or 2

  v8f acc[8];
#pragma unroll
  for (int n = 0; n < 8; ++n) acc[n] = (v8f)0.0f;

  // clamp A row for safety (M=100000 is a multiple of 16, so no-op here)
  int arow = mtile * 16 + sub;
  if (arow >= M) arow = M - 1;
  const float* __restrict__ xrow = x + (size_t)arow * IN_F;

  for (int k = 0; k < IN_F; k += 4) {
    // A fragment: two consecutive fp32 -> one b64 load
    v2f a = *(const v2f*)(xrow + k + khalf);
    const float* __restrict__ w0 = w + (size_t)(k + khalf) * OUT_F;
    const float* __restrict__ w1 = w0 + OUT_F;
#pragma unroll
    for (int n = 0; n < 8; ++n) {
      v2f b;
      b.x = w0[n * 16 + sub];
      b.y = w1[n * 16 + sub];
      acc[n] = __builtin_amdgcn_wmma_f32_16x16x4_f32(
          /*neg_a=*/false, a, /*neg_b=*/false, b,
          /*c_mod=*/(short)0, acc[n], /*reuse_a=*/false, /*reuse_b=*/false);
    }
  }

  // Store D: vgpr j holds row m = (lane>>4)*8 + j, column = (lane&15)+16n
  const int mbase = mtile * 16 + halfsel * 8;
#pragma unroll
  for (int j = 0; j < 8; ++j) {
    const int m = mbase + j;
    if (m < M) {
      float* __restrict__ dst = support + (size_t)m * OUT_F + sub;
#pragma unroll
      for (int n = 0; n < 8; ++n) dst[n * 16] = acc[n][j];
    }
  }
}

// -------------------------------------------------------------------------
// Kernel: out initialized to broadcast bias, float4 granularity (b128).
// 128 floats per row = 32 float4s; thread i writes out4[i] = bias4[i & 31].
// -------------------------------------------------------------------------
__global__ void __launch_bounds__(256)
gcn_init_out(float4* __restrict__ out4, const float4* __restrict__ bias4,
             int total4) {
  int i = (int)(blockIdx.x * 256u + threadIdx.x);
  if (i < total4) out4[i] = bias4[i & 31];
}

// -------------------------------------------------------------------------
extern "C" void kernel_launch(void* const* d_in, const int* in_sizes, int n_in,
                              void* d_out, int out_size, void* d_ws, size_t ws_size,
                              hipStream_t stream) {
  const float* x    = (const float*)d_in[0];
  const int*   erow = (const int*)  d_in[1];
  const int*   ecol = (const int*)  d_in[2];
  const float* eval = (const float*)d_in[3];
  const float* w    = (const float*)d_in[4];
  const float* bias = (const float*)d_in[5];
  float* out = (float*)d_out;
  float* support = (float*)d_ws;            // M*128 floats = 51.2 MB scratch

  const int M = in_sizes[0] / IN_F;         // 100000
  const int E = in_sizes[1];                // 3200000

  // 1) support = x @ W  (WMMA fp32)
  const int mtiles = (M + 15) / 16;         // 6250
  const int gemm_blocks = (mtiles + 7) / 8; // 8 waves (256 thr) per block
  gcn_gemm_wmma<<<gemm_blocks, 256, 0, stream>>>(x, w, support, M, mtiles);

  // 2) out = bias (broadcast), b128 stores
  const int total4 = M * (OUT_F / 4);
  gcn_init_out<<<(total4 + 255) / 256, 256, 0, stream>>>(
      (float4*)out, (const float4*)bias, total4);

  // 3) scatter-add: one wave per edge
  const long long nthreads = (long long)E * 32;
  const unsigned sblocks = (unsigned)((nthreads + 255) / 256);
  gcn_scatter<<<sblocks, 256, 0, stream>>>(support, erow, ecol, eval, out, E);
}